// TopkRouter_10239202034445
// MI455X (gfx1250) — compile-verified
//
#include <hip/hip_runtime.h>
#include <hip/hip_bf16.h>

typedef __attribute__((ext_vector_type(2))) float v2f;
typedef __attribute__((ext_vector_type(8))) float v8f;

#define TOPK 8
#define NEXP 256          // experts (r_b)
#define FOUT 1024         // output features
#define ROWS_PER_BLK 32
#define WD_STRIDE 260     // 256 + 4 pad -> LDS bank stride 4, conflict-free
#define NEG_INF (-3.402823466e38f)

__global__ __launch_bounds__(1024)
void topk_router_wmma(const float* __restrict__ Pb,
                      const float* __restrict__ V,
                      const float* __restrict__ bias,
                      float* __restrict__ out)
{
    __shared__ float Wd[ROWS_PER_BLK][WD_STRIDE];  // dense softmax weights
    __shared__ float b_s[NEXP];

    const int tid  = threadIdx.x;
    const int lane = tid & 31;
    const int wave = tid >> 5;                     // 0..31, one row each
    const int row0 = blockIdx.x * ROWS_PER_BLK;

    // zero dense weight matrix, stage bias
    for (int i = tid; i < ROWS_PER_BLK * WD_STRIDE; i += 1024)
        ((float*)Wd)[i] = 0.0f;
    if (tid < NEXP) b_s[tid] = bias[tid];
    __syncthreads();

    // ---------- Phase 1: per-wave top-8 + softmax ----------
    {
        const int row = row0 + wave;
        const float* prow = Pb + (size_t)row * NEXP;
        float sc[8];                                // bias-corrected scores
#pragma unroll
        for (int j = 0; j < 8; ++j) {
            int e = lane + 32 * j;
            sc[j] = prow[e] + b_s[e];
        }
        float cl[TOPK]; int ci[TOPK];
#pragma unroll
        for (int k = 0; k < TOPK; ++k) {
            // local argmax over this lane's 8 candidates
            float bv = sc[0]; int bi = lane;
#pragma unroll
            for (int j = 1; j < 8; ++j) {
                int   ii = lane + 32 * j;
                float v  = sc[j];
                if (v > bv) { bv = v; bi = ii; }
            }
            // wave32 butterfly argmax (tie -> lower index, like lax.top_k)
#pragma unroll
            for (int off = 16; off >= 1; off >>= 1) {
                float ov = __shfl_xor(bv, off, 32);
                int   oi = __shfl_xor(bi, off, 32);
                if (ov > bv || (ov == bv && oi < bi)) { bv = ov; bi = oi; }
            }
            ci[k] = bi;
            cl[k] = bv - b_s[bi];                   // recover ORIGINAL logit
            int jj = bi >> 5;
#pragma unroll
            for (int j = 0; j < 8; ++j)
                if (j == jj && lane == (bi & 31)) sc[j] = NEG_INF;
        }
        // softmax over chosen logits (values uniform across the wave)
        float mx = cl[0];
#pragma unroll
        for (int k = 1; k < TOPK; ++k) mx = fmaxf(mx, cl[k]);
        float s = 0.0f, w[TOPK];
#pragma unroll
        for (int k = 0; k < TOPK; ++k) { w[k] = __expf(cl[k] - mx); s += w[k]; }
        float inv = 1.0f / s;
        if (lane == 0) {
#pragma unroll
            for (int k = 0; k < TOPK; ++k)
                Wd[wave][ci[k]] = w[k] * inv;       // scatter into dense row
        }
    }
    __syncthreads();

    // ---------- Phase 2: out[32x1024] = Wd[32x256] @ V[256x1024] ----------
    const int m  = lane & 15;
    const int hi = lane >> 4;                       // 0: K{0,1}, 1: K{2,3}
#pragma unroll
    for (int t = 0; t < 4; ++t) {
        int T  = wave + 32 * t;                     // 0..127 wave-tiles
        int rt = T >> 6;                            // row tile (0..1)
        int ct = T & 63;                            // col tile (0..63)
        int col = ct * 16 + m;
        const float* vcol = V + col;
        const float* arow = &Wd[rt * 16 + m][0];
        v8f acc = {0.f,0.f,0.f,0.f,0.f,0.f,0.f,0.f};
        for (int k4 = 0; k4 < NEXP / 4; ++k4) {
            int kk = 4 * k4 + 2 * hi;
            v2f a, b;
            a.x = arow[kk];                         // A 16x4: M=lane&15
            a.y = arow[kk + 1];
            b.x = vcol[(size_t)kk * FOUT];          // B 4x16: N=lane&15
            b.y = vcol[(size_t)(kk + 1) * FOUT];
            acc = __builtin_amdgcn_wmma_f32_16x16x4_f32(
                      false, a, false, b, (short)0, acc, false, false);
        }
        // C/D layout: VGPR r -> M = r + 8*hi, N = m
        float* orow = out + (size_t)(row0 + rt * 16 + 8 * hi) * FOUT + col;
#pragma unroll
        for (int r = 0; r < 8; ++r)
            orow[(size_t)r * FOUT] = acc[r];
    }
}

extern "C" void kernel_launch(void* const* d_in, const int* in_sizes, int n_in,
                              void* d_out, int out_size, void* d_ws, size_t ws_size,
                              hipStream_t stream)
{
    (void)n_in; (void)out_size; (void)d_ws; (void)ws_size;
    const float* Pb   = (const float*)d_in[0];
    const float* V    = (const float*)d_in[1];
    const float* bias = (const float*)d_in[2];
    float* out = (float*)d_out;
    const int total_rows = in_sizes[0] / NEXP;      // 16384
    dim3 grid(total_rows / ROWS_PER_BLK);           // 512
    dim3 block(1024);                               // 32 wave32 waves
    hipLaunchKernelGGL(topk_router_wmma, grid, block, 0, stream,
                       Pb, V, bias, out);
}